// RotationR3Block_18846316495449
// MI455X (gfx1250) — compile-verified
//
#include <hip/hip_runtime.h>
#include <cstdint>
#include <cstddef>

#define NUM_QUBITS 10

typedef float v4f __attribute__((ext_vector_type(4)));

__device__ __forceinline__ float2 cmul(float2 a, float2 b) {
    return make_float2(a.x * b.x - a.y * b.y, a.x * b.y + a.y * b.x);
}

// One block per (batch b, high-row rh).  1024 threads = 32 waves; wave w owns
// output row r = rh*32 + w.  Each thread streams 16 float4 NT stores.
__global__ __launch_bounds__(1024) void kron_rows_kernel(
    const float* __restrict__ angles, float* __restrict__ out) {

    const int tid = threadIdx.x;
    const int b   = blockIdx.x >> 5;
    const int rh  = blockIdx.x & 31;

    __shared__ float  sAng[32];    // 30 angle floats for this batch
    __shared__ float2 sMat[40];    // 10 matrices x 4 complex entries
    __shared__ float2 sHi[32];     // Hi[rh, ch], ch = 0..31
    __shared__ float2 sLo[1024];   // Lo[rl, cl]

    // ---- CDNA5 async global->LDS copy of the angle block (120 B) ----------
    if (tid < NUM_QUBITS * 3) {
        uint32_t lds_addr = (uint32_t)(size_t)(&sAng[tid]);   // low 32b = LDS offset
        uint32_t voff     = (uint32_t)tid * 4u;
        const float* base = angles + b * (NUM_QUBITS * 3);
        asm volatile("global_load_async_to_lds_b32 %0, %1, %2 offset:0"
                     :: "v"(lds_addr), "v"(voff), "s"(base)
                     : "memory");
    }
    asm volatile("s_wait_asynccnt 0" ::: "memory");
    __syncthreads();

    // ---- build the 10 single-qubit 2x2 matrices ---------------------------
    if (tid < NUM_QUBITS) {
        float omega = sAng[tid * 3 + 0];
        float theta = sAng[tid * 3 + 1];
        float phi   = sAng[tid * 3 + 2];
        float sh, chh;
        sincosf(theta * 0.5f, &sh, &chh);
        float a = (phi + omega) * 0.5f;
        float d = (phi - omega) * 0.5f;
        float sa, ca, sd, cd;
        sincosf(a, &sa, &ca);
        sincosf(d, &sd, &cd);
        sMat[tid * 4 + 0] = make_float2( ca * chh, -sa * chh);  // e^{-ia} cos
        sMat[tid * 4 + 1] = make_float2(-cd * sh,  -sd * sh );  // -e^{+id} sin
        sMat[tid * 4 + 2] = make_float2( cd * sh,  -sd * sh );  // e^{-id} sin
        sMat[tid * 4 + 3] = make_float2( ca * chh,  sa * chh);  // e^{+ia} cos
    }
    __syncthreads();

    // ---- partial Kronecker factors ---------------------------------------
    {   // Lo[rl, cl] = prod_{q=5..9} M_q ;  qubit 5+q controls bit (4-q)
        int rl = tid >> 5, cl = tid & 31;
        float2 p = make_float2(1.0f, 0.0f);
        #pragma unroll
        for (int q = 0; q < 5; ++q) {
            int rb = (rl >> (4 - q)) & 1;
            int cb = (cl >> (4 - q)) & 1;
            p = cmul(p, sMat[(5 + q) * 4 + rb * 2 + cb]);
        }
        sLo[tid] = p;
    }
    if (tid < 32) {  // Hi[rh, ch] row for this block
        int chc = tid;
        float2 p = make_float2(1.0f, 0.0f);
        #pragma unroll
        for (int q = 0; q < 5; ++q) {
            int rb = (rh  >> (4 - q)) & 1;
            int cb = (chc >> (4 - q)) & 1;
            p = cmul(p, sMat[q * 4 + rb * 2 + cb]);
        }
        sHi[tid] = p;
    }
    __syncthreads();

    // ---- stream one 8 KB output row per wave ------------------------------
    const int lane = tid & 31;
    const int rl   = tid >> 5;
    const int r    = rh * 32 + rl;

    // each lane keeps its two Lo values in registers for the whole row
    const float2 lo0 = sLo[rl * 32 + (lane & 15) * 2 + 0];
    const float2 lo1 = sLo[rl * 32 + (lane & 15) * 2 + 1];

    float* rowp = out + (((size_t)b * 1024 + (size_t)r) * 1024) * 2;

    #pragma unroll
    for (int j = 0; j < 16; ++j) {
        // complex cols c0 = j*64 + 2*lane, c0+1 ; ch = 2j + (lane>=16)
        float2 h  = sHi[2 * j + (lane >> 4)];
        float2 o0 = cmul(h, lo0);
        float2 o1 = cmul(h, lo1);
        v4f v;
        v.x = o0.x; v.y = o0.y; v.z = o1.x; v.w = o1.y;
        int c0 = j * 64 + lane * 2;
        __builtin_nontemporal_store(v, (v4f*)(rowp + (size_t)c0 * 2));
    }
}

extern "C" void kernel_launch(void* const* d_in, const int* in_sizes, int n_in,
                              void* d_out, int out_size, void* d_ws, size_t ws_size,
                              hipStream_t stream) {
    (void)n_in; (void)d_ws; (void)ws_size; (void)out_size;
    const float* angles = (const float*)d_in[0];
    float* out = (float*)d_out;
    int B = in_sizes[0] / (NUM_QUBITS * 3);   // 32 for the reference
    dim3 grid((unsigned)(B * 32)), block(1024);
    hipLaunchKernelGGL(kron_rows_kernel, grid, block, 0, stream, angles, out);
}